// DCCF_87900800680713
// MI455X (gfx1250) — compile-verified
//
#include <hip/hip_runtime.h>

// ---- problem constants (fixed by setup_inputs) ----
constexpr int N_USERS = 30000;
constexpr int N_ITEMS = 60000;
constexpr int NN      = N_USERS + N_ITEMS;   // 90000
constexpr int D       = 64;
constexpr int BATCH   = 8192;
constexpr int NLAYERS = 2;
constexpr int TPB16   = BATCH / 16;          // 512 tiles per matrix dim
constexpr int CHUNK   = 32;                  // tn tiles processed per block
constexpr int TMGRP   = TPB16 / 8;           // 64 groups of 8 tm-tiles (one per wave)

constexpr float TEMP_INV  = 5.0f;            // 1/0.2
constexpr float EMB_REG_C = 1e-05f;
constexpr float SSL_REG_C = 0.1f;
constexpr float EPS_PERT  = 0.2f;

typedef __attribute__((ext_vector_type(2))) float v2f;
typedef __attribute__((ext_vector_type(8))) float v8f;

// GCC-style int4 vector + address-space-qualified pointers for the async builtin:
// signature is (v4i AS1* src_global, v4i AS3* dst_lds, imm offset, imm cpol)
typedef int v4i __attribute__((vector_size(16)));
typedef v4i __attribute__((address_space(1))) * gas_v4i;
typedef v4i __attribute__((address_space(3))) * las_v4i;

#if __has_builtin(__builtin_amdgcn_global_load_async_to_lds_b128)
#define USE_ASYNC_LDS 1
#endif

// ---------------- degree / d_inv ----------------
__global__ void deg_kernel(const int* __restrict__ h, float* __restrict__ deg, int E) {
    int i = blockIdx.x * blockDim.x + threadIdx.x;
    if (i < E) atomicAdd(&deg[h[i]], 1.0f);
}

__global__ void dinv_kernel(float* __restrict__ dinv, int n) {
    int i = blockIdx.x * blockDim.x + threadIdx.x;
    if (i < n) {
        float d = dinv[i];
        dinv[i] = rsqrtf(d > 0.0f ? d : 1.0f);
    }
}

// ---------------- e0 broadcast into 3 stream buffers + 3 accumulators ----------------
__global__ void init_streams(const float* __restrict__ ue, const float* __restrict__ ie,
                             float* c0, float* c1, float* c2,
                             float* a0, float* a1, float* a2) {
    int i = blockIdx.x * blockDim.x + threadIdx.x;
    if (i >= NN * D) return;
    int row = i >> 6, col = i & 63;
    float v = (row < N_USERS) ? ue[row * D + col] : ie[(row - N_USERS) * D + col];
    c0[i] = v; c1[i] = v; c2[i] = v;
    a0[i] = v; a1[i] = v; a2[i] = v;
}

// ---------------- SpMM: y[h[e]] += dinv[h]*dinv[t] * x[t[e]] ----------------
// 16 threads per edge, float4 per thread; operands L2-resident (23 MB each vs 192 MB L2).
__global__ void spmm_kernel(const float* __restrict__ x, float* __restrict__ y,
                            const int* __restrict__ h, const int* __restrict__ t,
                            const float* __restrict__ dinv, int E) {
    int tid = blockIdx.x * blockDim.x + threadIdx.x;
    int e = tid >> 4, seg = tid & 15;
    if (e >= E) return;
    int hh = h[e], tt = t[e];
    float g = dinv[hh] * dinv[tt];
    float4 v = reinterpret_cast<const float4*>(x + (size_t)tt * D)[seg];
    float* yp = y + (size_t)hh * D + seg * 4;
    atomicAdd(yp + 0, g * v.x);
    atomicAdd(yp + 1, g * v.y);
    atomicAdd(yp + 2, g * v.z);
    atomicAdd(yp + 3, g * v.w);
}

// ---------------- plain stream: cur = nxt; acc += nxt ----------------
__global__ void plain_acc(float* __restrict__ cur, float* __restrict__ acc,
                          const float* __restrict__ nxt, int n) {
    int i = blockIdx.x * blockDim.x + threadIdx.x;
    if (i < n) {
        float v = nxt[i];
        cur[i] = v;
        acc[i] += v;
    }
}

// ---------------- perturbed stream: cur = nxt + rownorm(noise)*sign(nxt)*EPS; acc += cur ----------------
__global__ void perturb_acc(float* __restrict__ cur, float* __restrict__ acc,
                            const float* __restrict__ nxt, const float* __restrict__ nz,
                            int nrows) {
    int wid  = (blockIdx.x * blockDim.x + threadIdx.x) >> 5;
    int lane = threadIdx.x & 31;
    if (wid >= nrows) return;
    size_t base = (size_t)wid * D;
    float n0 = nz[base + lane], n1 = nz[base + lane + 32];
    float ss = n0 * n0 + n1 * n1;
    for (int m = 16; m >= 1; m >>= 1) ss += __shfl_xor(ss, m, 32);
    float rn = 1.0f / fmaxf(sqrtf(ss), 1e-12f);
#pragma unroll
    for (int j = 0; j < 2; ++j) {
        size_t idx = base + lane + j * 32;
        float v  = nxt[idx];
        float sg = (v > 0.0f) ? 1.0f : ((v < 0.0f) ? -1.0f : 0.0f);
        float p  = v + (j ? n1 : n0) * rn * sg * EPS_PERT;
        cur[idx] = p;
        acc[idx] += p;
    }
}

// ---------------- BPR + embedding regularizer (warp per batch element) ----------------
__global__ void bpr_emb_kernel(const int* __restrict__ users, const int* __restrict__ pos,
                               const int* __restrict__ neg, const float* __restrict__ acc0,
                               const float* __restrict__ ue, const float* __restrict__ ie,
                               float* __restrict__ out) {
    int wid  = (blockIdx.x * blockDim.x + threadIdx.x) >> 5;
    int lane = threadIdx.x & 31;
    if (wid >= BATCH) return;
    int u = users[wid], p = pos[wid], n = neg[wid];
    const float* ur = acc0 + (size_t)u * D;
    const float* pr = acc0 + (size_t)(N_USERS + p) * D;
    const float* nr = acc0 + (size_t)(N_USERS + n) * D;
    const float* eu = ue + (size_t)u * D;
    const float* ep = ie + (size_t)p * D;
    const float* en = ie + (size_t)n * D;
    float dp = 0.f, dn = 0.f, sq = 0.f;
#pragma unroll
    for (int j = 0; j < 2; ++j) {
        int k = lane + j * 32;
        float uv = ur[k];
        dp += uv * pr[k];
        dn += uv * nr[k];
        float a = eu[k], b = ep[k], c = en[k];
        sq += a * a + b * b + c * c;
    }
    for (int m = 16; m >= 1; m >>= 1) {
        dp += __shfl_xor(dp, m, 32);
        dn += __shfl_xor(dn, m, 32);
        sq += __shfl_xor(sq, m, 32);
    }
    if (lane == 0) {
        float x  = dn - dp;
        float sp = fmaxf(x, 0.0f) + log1pf(expf(-fabsf(x)));   // stable softplus
        atomicAdd(out + 0, sp / (float)BATCH);
        atomicAdd(out + 3, EMB_REG_C * sq);
    }
}

// ---------------- gather + row-normalize the 4 contrastive views, pos diagonal ----------------
__global__ void gather_norm_kernel(const int* __restrict__ users, const int* __restrict__ pos,
                                   const float* __restrict__ acc1, const float* __restrict__ acc2,
                                   float* __restrict__ u1, float* __restrict__ u2,
                                   float* __restrict__ i1, float* __restrict__ i2,
                                   float* __restrict__ pos_u, float* __restrict__ pos_i) {
    int wid  = (blockIdx.x * blockDim.x + threadIdx.x) >> 5;
    int lane = threadIdx.x & 31;
    if (wid >= BATCH) return;
    int u = users[wid], p = pos[wid];
    const float* rows[4] = { acc1 + (size_t)u * D, acc2 + (size_t)u * D,
                             acc1 + (size_t)(N_USERS + p) * D, acc2 + (size_t)(N_USERS + p) * D };
    float* outs[4] = { u1 + (size_t)wid * D, u2 + (size_t)wid * D,
                       i1 + (size_t)wid * D, i2 + (size_t)wid * D };
    float v[4][2];
#pragma unroll
    for (int r = 0; r < 4; ++r) {
        float a = rows[r][lane], b = rows[r][lane + 32];
        float ss = a * a + b * b;
        for (int m = 16; m >= 1; m >>= 1) ss += __shfl_xor(ss, m, 32);
        float rn = 1.0f / fmaxf(sqrtf(ss), 1e-12f);
        v[r][0] = a * rn; v[r][1] = b * rn;
        outs[r][lane]      = v[r][0];
        outs[r][lane + 32] = v[r][1];
    }
    float du = v[0][0] * v[1][0] + v[0][1] * v[1][1];
    float di = v[2][0] * v[3][0] + v[2][1] * v[3][1];
    for (int m = 16; m >= 1; m >>= 1) {
        du += __shfl_xor(du, m, 32);
        di += __shfl_xor(di, m, 32);
    }
    if (lane == 0) {
        pos_u[wid] = __expf(du * TEMP_INV);
        pos_i[wid] = __expf(di * TEMP_INV);
    }
}

// ---------------- neg[b] = sum_c exp((A[b]·Bm[c]) / TEMP) via f32 WMMA ----------------
// Block = 8 waves; each wave owns one tm tile (A frags register-resident, 32 VGPRs)
// and iterates over a CHUNK of tn tiles. The 4 KB B tile is staged into LDS once per
// block via async Global->LDS DMA (ASYNCcnt path) and shared by all 8 waves.
// exp row-sums accumulate in registers across the chunk -> one atomicAdd per row per wave.
__global__ void __launch_bounds__(256)
wmma_neg_kernel(const float* __restrict__ A, const float* __restrict__ Bm,
                float* __restrict__ neg) {
    __shared__ float Btile[16 * D];          // 4 KB, one 16-row B panel
    const int lane  = threadIdx.x & 31;
    const int wave  = threadIdx.x >> 5;
    const int tm    = (blockIdx.x % TMGRP) * 8 + wave;
    const int chunk = blockIdx.x / TMGRP;
    const int half  = lane >> 4;
    const int l16   = lane & 15;

    // preload the wave's A fragments for all 16 K-steps (K=64, 4 per WMMA)
    const float* arow = A + (size_t)(tm * 16 + l16) * D + 2 * half;
    v2f afrag[16];
#pragma unroll
    for (int k = 0; k < 16; ++k) afrag[k] = v2f{ arow[4 * k], arow[4 * k + 1] };

    float rowsum[8];
#pragma unroll
    for (int r = 0; r < 8; ++r) rowsum[r] = 0.0f;

    for (int c = 0; c < CHUNK; ++c) {
        const int tn = chunk * CHUNK + c;
        const float* bsrc = Bm + (size_t)tn * 16 * D;
        __syncthreads();                     // previous tile fully consumed
#ifdef USE_ASYNC_LDS
        // 256 threads x 16B = 4 KB: DMA straight into LDS, no VGPR staging
        __builtin_amdgcn_global_load_async_to_lds_b128(
            (gas_v4i)(bsrc + threadIdx.x * 4),
            (las_v4i)(Btile + threadIdx.x * 4),
            0, 0);
#if __has_builtin(__builtin_amdgcn_s_wait_asynccnt)
        __builtin_amdgcn_s_wait_asynccnt(0);
#else
        asm volatile("s_wait_asynccnt 0x0" ::: "memory");
#endif
#else
        float4 tv = reinterpret_cast<const float4*>(bsrc)[threadIdx.x];
        reinterpret_cast<float4*>(Btile)[threadIdx.x] = tv;
#endif
        __syncthreads();

        v8f cacc = {};
#pragma unroll
        for (int k = 0; k < 16; ++k) {
            const float* bp = Btile + l16 * D + 4 * k + 2 * half;
            v2f b = v2f{ bp[0], bp[1] };
            cacc = __builtin_amdgcn_wmma_f32_16x16x4_f32(false, afrag[k], false, b,
                                                         (short)0, cacc, false, false);
        }
#pragma unroll
        for (int r = 0; r < 8; ++r) rowsum[r] += __expf(cacc[r] * TEMP_INV);
    }

    // D layout: VGPR r, lanes 0-15 -> (M=r, N=l16); lanes 16-31 -> (M=r+8, N=l16)
#pragma unroll
    for (int r = 0; r < 8; ++r) {
        float e = rowsum[r];
        for (int m = 8; m >= 1; m >>= 1) e += __shfl_xor(e, m, 32);
        if (l16 == 0) atomicAdd(&neg[tm * 16 + r + 8 * half], e);
    }
}

// ---------------- final contrastive reduction ----------------
__global__ void cse_final_kernel(const float* __restrict__ pos_u, const float* __restrict__ neg_u,
                                 const float* __restrict__ pos_i, const float* __restrict__ neg_i,
                                 float* __restrict__ out) {
    int i = blockIdx.x * blockDim.x + threadIdx.x;
    if (i >= 2 * BATCH) return;
    const float* P = (i < BATCH) ? pos_u : pos_i;
    const float* Q = (i < BATCH) ? neg_u : neg_i;
    int b = (i < BATCH) ? i : i - BATCH;
    float l = -logf(P[b] / (Q[b] + 1e-8f) + 1e-8f);
    atomicAdd(out + 2, SSL_REG_C * l / (float)BATCH);
}

extern "C" void kernel_launch(void* const* d_in, const int* in_sizes, int n_in,
                              void* d_out, int out_size, void* d_ws, size_t ws_size,
                              hipStream_t stream) {
    const int*   users     = (const int*)d_in[0];
    const int*   pos_items = (const int*)d_in[1];
    const int*   neg_items = (const int*)d_in[2];
    const float* user_emb  = (const float*)d_in[3];
    const float* item_emb  = (const float*)d_in[4];
    const int*   h_list    = (const int*)d_in[5];
    const int*   t_list    = (const int*)d_in[6];
    const float* noise     = (const float*)d_in[7];
    const int E = in_sizes[5];
    float* out = (float*)d_out;

    // workspace layout (floats)
    float* ws = (float*)d_ws;
    size_t off = 0;
    auto alloc = [&](size_t n) { float* p = ws + off; off += n; return p; };
    float* dinv   = alloc(NN);
    float* cur[3] = { alloc((size_t)NN * D), alloc((size_t)NN * D), alloc((size_t)NN * D) };
    float* acc[3] = { alloc((size_t)NN * D), alloc((size_t)NN * D), alloc((size_t)NN * D) };
    float* nxt    = alloc((size_t)NN * D);
    float* u1 = alloc((size_t)BATCH * D);
    float* u2 = alloc((size_t)BATCH * D);
    float* i1 = alloc((size_t)BATCH * D);
    float* i2 = alloc((size_t)BATCH * D);
    float* pos_u = alloc(BATCH);
    float* pos_i = alloc(BATCH);
    float* neg_u = alloc(BATCH);
    float* neg_i = alloc(BATCH);

    (void)hipMemsetAsync(out, 0, 4 * sizeof(float), stream);      // also zeroes svd slot
    (void)hipMemsetAsync(dinv, 0, NN * sizeof(float), stream);
    deg_kernel<<<(E + 255) / 256, 256, 0, stream>>>(h_list, dinv, E);
    dinv_kernel<<<(NN + 255) / 256, 256, 0, stream>>>(dinv, NN);
    init_streams<<<(NN * D + 255) / 256, 256, 0, stream>>>(user_emb, item_emb,
                                                           cur[0], cur[1], cur[2],
                                                           acc[0], acc[1], acc[2]);

    for (int l = 0; l < NLAYERS; ++l) {
        for (int s = 0; s < 3; ++s) {
            (void)hipMemsetAsync(nxt, 0, (size_t)NN * D * sizeof(float), stream);
            long tids = (long)E * 16;
            spmm_kernel<<<(int)((tids + 255) / 256), 256, 0, stream>>>(cur[s], nxt, h_list, t_list, dinv, E);
            if (s == 0) {
                plain_acc<<<(NN * D + 255) / 256, 256, 0, stream>>>(cur[0], acc[0], nxt, NN * D);
            } else {
                const float* nz = noise + (size_t)(l * 2 + (s - 1)) * NN * D;
                perturb_acc<<<(NN * 32 + 255) / 256, 256, 0, stream>>>(cur[s], acc[s], nxt, nz, NN);
            }
        }
    }

    bpr_emb_kernel<<<(BATCH * 32) / 256, 256, 0, stream>>>(users, pos_items, neg_items,
                                                           acc[0], user_emb, item_emb, out);
    gather_norm_kernel<<<(BATCH * 32) / 256, 256, 0, stream>>>(users, pos_items, acc[1], acc[2],
                                                               u1, u2, i1, i2, pos_u, pos_i);
    (void)hipMemsetAsync(neg_u, 0, BATCH * sizeof(float), stream);
    (void)hipMemsetAsync(neg_i, 0, BATCH * sizeof(float), stream);
    const int blocks = TMGRP * (TPB16 / CHUNK);   // 64 * 16 = 1024 blocks of 8 waves
    wmma_neg_kernel<<<blocks, 256, 0, stream>>>(u1, u2, neg_u);
    wmma_neg_kernel<<<blocks, 256, 0, stream>>>(i1, i2, neg_i);
    cse_final_kernel<<<(2 * BATCH + 255) / 256, 256, 0, stream>>>(pos_u, neg_u, pos_i, neg_i, out);
}